// SoftPatternClassifier_55293408968804
// MI455X (gfx1250) — compile-verified
//
#include <hip/hip_runtime.h>
#include <hip/hip_bf16.h>

// ---------------------------------------------------------------------------
// SoPa (SoftPatternClassifier) forward for MI455X / gfx1250 (wave32, WMMA).
//   dims: B=32, T=128, D=300, V=50000, P=512, ND=2, MAXL=7, K=P*ND*MAXL=7168
//   pipeline:
//     k0: transpose diags [7168,300] -> K-pair-interleaved [150][7168][2]
//     k1: WMMA f32 GEMM  ts[4096,7168] = gather(emb_table,tokens) . diags^T + bias
//         wave tile 32(M) x 64(N): B-fragments reused across 2 M-tiles
//     k2: max-sum semiring scan over T per (b,p) -> scores[32,512]
//     k3: batchnorm(batch stats) + binarize + final linear -> out[32,2]
// ---------------------------------------------------------------------------

#define B_   32
#define T_   128
#define D_   300
#define P_   512
#define KTOT 7168          // P * ND * MAXL
#define M_   4096          // B * T
#define NEGF (-1.0e30f)

typedef __attribute__((ext_vector_type(2))) float v2f;
typedef __attribute__((ext_vector_type(8))) float v8f;

// ---------------- kernel 0: diag transpose into K-pair-major layout ---------
// diagsP[((k>>1)*KTOT + n)*2 + (k&1)] = diags[n*D + k]
__global__ __launch_bounds__(256) void k0_transpose_diags(
    const float* __restrict__ diags, float* __restrict__ diagsP)
{
    int tid = blockIdx.x * 256 + threadIdx.x;
    if (tid >= KTOT * D_) return;
    int n = tid / D_;
    int k = tid - n * D_;
    diagsP[((size_t)(k >> 1) * KTOT + n) * 2 + (k & 1)] = diags[tid];
}

// ---------------- kernel 1: fused gather + f32 WMMA GEMM + bias -------------
// grid (112, 16), block 256 (8 waves). Wave w covers M-tiles {base, base+1}
// (32 rows) and a 64-col N strip (4 x 16). 75 k-steps of K=4, 8 WMMAs each.
__global__ __launch_bounds__(256) void k1_gemm_ts(
    const int*   __restrict__ tokens,
    const float* __restrict__ emb_table,
    const float* __restrict__ diagsP,
    const float* __restrict__ bias,
    float*       __restrict__ ts)
{
    const int lane = threadIdx.x & 31;
    const int wave = threadIdx.x >> 5;
    const int half = lane >> 4;          // 0: K={k,k+1}, 1: K={k+2,k+3}
    const int l16  = lane & 15;

    const int mt0 = (blockIdx.y * 8 + wave) * 2;  // first of two M tiles
    const int n0  = blockIdx.x * 64;              // N strip base
    const int row0 = mt0 * 16 + l16;              // lane's row in tile 0
    const int row1 = row0 + 16;                   // lane's row in tile 1

    const float* abase0 = emb_table + (size_t)tokens[row0] * D_;
    const float* abase1 = emb_table + (size_t)tokens[row1] * D_;
    const int    col    = n0 + l16;

    v8f c00 = {}, c01 = {}, c02 = {}, c03 = {};   // M-tile 0, N-subtiles 0..3
    v8f c10 = {}, c11 = {}, c12 = {}, c13 = {};   // M-tile 1, N-subtiles 0..3

    for (int k = 0; k < D_; k += 4) {
        const int kk = k + 2 * half;
        v2f a0 = *(const v2f*)(abase0 + kk);                     // K={kk,kk+1}
        v2f a1 = *(const v2f*)(abase1 + kk);
        const float* bp = diagsP + ((size_t)(kk >> 1) * KTOT + col) * 2;
        v2f b0 = *(const v2f*)(bp);
        v2f b1 = *(const v2f*)(bp + 32);                         // +16 cols
        v2f b2 = *(const v2f*)(bp + 64);
        v2f b3 = *(const v2f*)(bp + 96);
        c00 = __builtin_amdgcn_wmma_f32_16x16x4_f32(false, a0, false, b0,
                                                    (short)0, c00, false, false);
        c10 = __builtin_amdgcn_wmma_f32_16x16x4_f32(false, a1, false, b0,
                                                    (short)0, c10, false, false);
        c01 = __builtin_amdgcn_wmma_f32_16x16x4_f32(false, a0, false, b1,
                                                    (short)0, c01, false, false);
        c11 = __builtin_amdgcn_wmma_f32_16x16x4_f32(false, a1, false, b1,
                                                    (short)0, c11, false, false);
        c02 = __builtin_amdgcn_wmma_f32_16x16x4_f32(false, a0, false, b2,
                                                    (short)0, c02, false, false);
        c12 = __builtin_amdgcn_wmma_f32_16x16x4_f32(false, a1, false, b2,
                                                    (short)0, c12, false, false);
        c03 = __builtin_amdgcn_wmma_f32_16x16x4_f32(false, a0, false, b3,
                                                    (short)0, c03, false, false);
        c13 = __builtin_amdgcn_wmma_f32_16x16x4_f32(false, a1, false, b3,
                                                    (short)0, c13, false, false);
    }

    // C/D layout: VGPR r, lanes 0-15 -> M=r, lanes 16-31 -> M=8+r; N = lane&15
    const int mbase0 = mt0 * 16 + 8 * half;
    const int mbase1 = mbase0 + 16;
    #pragma unroll
    for (int j = 0; j < 4; ++j) {
        const int cj = n0 + j * 16 + l16;
        const float bv = bias[cj];
        v8f ca = (j == 0) ? c00 : (j == 1) ? c01 : (j == 2) ? c02 : c03;
        v8f cb = (j == 0) ? c10 : (j == 1) ? c11 : (j == 2) ? c12 : c13;
        float* orow0 = ts + (size_t)mbase0 * KTOT + cj;
        float* orow1 = ts + (size_t)mbase1 * KTOT + cj;
        #pragma unroll
        for (int r = 0; r < 8; ++r) {
            orow0[(size_t)r * KTOT] = ca[r] + bv;
            orow1[(size_t)r * KTOT] = cb[r] + bv;
        }
    }
}

// ---------------- kernel 2: max-sum semiring scan over T --------------------
// one thread per (b,p); ts slice per step = 14 consecutive floats at p*14.
__global__ __launch_bounds__(256) void k2_scan(
    const float* __restrict__ ts,
    const int*   __restrict__ doc_lens,
    const float* __restrict__ epsilons,
    float*       __restrict__ scores)
{
    const int tid = blockIdx.x * 256 + threadIdx.x;   // 0..16383
    const int b = tid >> 9;
    const int p = tid & (P_ - 1);

    float eps[6];
    #pragma unroll
    for (int i = 0; i < 6; ++i) eps[i] = epsilons[p * 6 + i];

    const int dl   = doc_lens[b];
    const int endi = (p < 256) ? 5 : 6;               // END_STATES

    float hid[7];
    hid[0] = 0.0f;
    #pragma unroll
    for (int l = 1; l < 7; ++l) hid[l] = NEGF;
    float sc = NEGF;

    const float* base = ts + (size_t)b * T_ * KTOT + p * 14;

    for (int t = 0; t < T_; ++t) {
        const float2* tp = (const float2*)(base + (size_t)t * KTOT);
        float v[14];
        #pragma unroll
        for (int i = 0; i < 7; ++i) {
            float2 x = tp[i];
            v[2 * i] = x.x; v[2 * i + 1] = x.y;
        }
        // epsilon transitions: ae[l] = max(hid[l], hid[l-1] + eps[l-1])
        float ae[7];
        ae[0] = hid[0];
        #pragma unroll
        for (int l = 1; l < 7; ++l) ae[l] = fmaxf(hid[l], hid[l - 1] + eps[l - 1]);
        // hid[l] = max(main, self-loop); main[0] = restart padding (0)
        float nh[7];
        nh[0] = fmaxf(0.0f, ae[0] + v[0]);
        #pragma unroll
        for (int l = 1; l < 7; ++l)
            nh[l] = fmaxf(ae[l - 1] + v[7 + l - 1], ae[l] + v[l]);
        #pragma unroll
        for (int l = 0; l < 7; ++l) hid[l] = nh[l];
        if (t < dl) sc = fmaxf(sc, hid[endi]);
    }
    scores[b * P_ + p] = sc;
}

// ---------------- kernel 3: batchnorm + binarize + final linear -------------
// single block of 512 threads (thread = pattern p). Deterministic: binarized
// values go into an LDS bitmask (integer atomics), then an ordered sum.
__global__ __launch_bounds__(512) void k3_finalize(
    const float* __restrict__ scores,
    const float* __restrict__ bn_w, const float* __restrict__ bn_b,
    const float* __restrict__ fw,   const float* __restrict__ fb,
    float* __restrict__ out)
{
    __shared__ unsigned int bits[B_ * 16];    // 512 bits per batch row
    const int p = threadIdx.x;
    bits[p] = 0u;
    __syncthreads();

    float s[B_];
    float m = 0.0f;
    #pragma unroll
    for (int b = 0; b < B_; ++b) { s[b] = scores[b * P_ + p]; m += s[b]; }
    m *= (1.0f / B_);
    float var = 0.0f;
    #pragma unroll
    for (int b = 0; b < B_; ++b) { float d = s[b] - m; var += d * d; }
    var *= (1.0f / B_);

    const float inv = rsqrtf(var + 1e-5f) * bn_w[p];
    const float bb  = bn_b[p];
    #pragma unroll
    for (int b = 0; b < B_; ++b) {
        float z = (s[b] - m) * inv + bb;      // sign(relu(z)) == (z > 0)
        if (z > 0.0f)
            atomicOr(&bits[b * 16 + (p >> 5)], 1u << (p & 31));
    }
    __syncthreads();

    if (p < B_ * 2) {
        const int b = p >> 1, c = p & 1;
        float sum = 0.0f;
        for (int q = 0; q < P_; ++q)
            if ((bits[b * 16 + (q >> 5)] >> (q & 31)) & 1u)
                sum += fw[c * P_ + q];
        out[b * 2 + c] = sum + fb[c];
    }
}

// ---------------------------------------------------------------------------
extern "C" void kernel_launch(void* const* d_in, const int* in_sizes, int n_in,
                              void* d_out, int out_size, void* d_ws, size_t ws_size,
                              hipStream_t stream) {
    const int*   tokens   = (const int*)  d_in[0];
    const int*   doc_lens = (const int*)  d_in[1];
    const float* emb      = (const float*)d_in[2];
    const float* diags    = (const float*)d_in[3];
    const float* bias     = (const float*)d_in[4];
    const float* eps      = (const float*)d_in[5];
    const float* bnw      = (const float*)d_in[6];
    const float* bnb      = (const float*)d_in[7];
    const float* fw       = (const float*)d_in[8];
    const float* fb       = (const float*)d_in[9];
    float* out = (float*)d_out;

    char* ws = (char*)d_ws;
    const size_t diagsP_bytes = (size_t)(D_ / 2) * KTOT * 2 * sizeof(float); // 8.6 MB
    const size_t ts_bytes     = (size_t)M_ * KTOT * sizeof(float);           // 117.4 MB
    float* diagsP = (float*)ws;
    float* ts     = (float*)(ws + diagsP_bytes);
    float* scores = (float*)(ws + diagsP_bytes + ts_bytes);

    {   // k0: 7168*300 elements
        int total = KTOT * D_;
        k0_transpose_diags<<<(total + 255) / 256, 256, 0, stream>>>(diags, diagsP);
    }
    {   // k1: N strips of 64 (112) x M pairs of tiles: 4096/(32*8) = 16
        dim3 grid(KTOT / 64, M_ / (32 * 8));
        k1_gemm_ts<<<grid, 256, 0, stream>>>(tokens, emb, diagsP, bias, ts);
    }
    {   // k2: one thread per (b,p)
        k2_scan<<<(B_ * P_) / 256, 256, 0, stream>>>(ts, doc_lens, eps, scores);
    }
    {   // k3: single block
        k3_finalize<<<1, 512, 0, stream>>>(scores, bnw, bnb, fw, fb, out);
    }
}